// ResidualVectorQuantizer_61984968016344
// MI455X (gfx1250) — compile-verified
//
#include <hip/hip_runtime.h>

typedef float v2f __attribute__((ext_vector_type(2)));
typedef float v8f __attribute__((ext_vector_type(8)));

#define N_Q    4
#define N_E    2048
#define E_DIM  256
#define N_TOK  (16 * 2048)          // 32768 tokens
#define TPW    32                   // tokens per wave (two 16-row WMMA tiles)
#define WPW    4                    // waves per workgroup
#define LDS_STRIDE 260              // 260 % 64 == 4 -> conflict-free float2 ds loads
#define ZQ_SIZE ((size_t)N_TOK * E_DIM)
#define NUM_TILES (N_TOK / TPW)     // 1024 waves total

// ---------------------------------------------------------------------------
// Kernel 1: per-code squared norms, one wave per codebook row (coalesced).
// ---------------------------------------------------------------------------
__global__ void rvq_norms_kernel(const float* __restrict__ W,
                                 float* __restrict__ norms) {
    const int row  = blockIdx.x * 8 + (threadIdx.x >> 5);   // 0 .. N_Q*N_E-1
    const int lane = threadIdx.x & 31;
    const float* wrow = W + (size_t)row * E_DIM;
    float s = 0.f;
    for (int d = lane; d < E_DIM; d += 32) {
        float v = wrow[d];
        s = fmaf(v, v, s);
    }
    #pragma unroll
    for (int m = 16; m >= 1; m >>= 1) s += __shfl_xor(s, m, 32);
    if (lane == 0) norms[row] = s;
}

// ---------------------------------------------------------------------------
// Kernel 2: main residual-VQ. Each wave owns 32 tokens resident in LDS and
// runs all 4 codebook stages: WMMA distance GEMM -> argmin -> gather/update.
// ---------------------------------------------------------------------------
__global__ void __launch_bounds__(WPW * 32)
rvq_main_kernel(const float* __restrict__ z, const float* __restrict__ W,
                const float* __restrict__ norms, float* __restrict__ zq,
                float* __restrict__ idx_out, float* __restrict__ loss_partial) {
    extern __shared__ float smem[];
    const int wave = threadIdx.x >> 5;
    const int lane = threadIdx.x & 31;
    const int half = lane >> 4;        // 0: lanes 0-15, 1: lanes 16-31
    const int l16  = lane & 15;

    float* R    = smem + wave * (TPW * LDS_STRIDE);                  // [32][260]
    int*   widx = (int*)(smem + WPW * TPW * LDS_STRIDE) + wave * TPW;

    const int    tile = blockIdx.x * WPW + wave;   // 0 .. NUM_TILES-1
    const size_t tok0 = (size_t)tile * TPW;

    // Load z tile into LDS residual.
    for (int t = 0; t < TPW; ++t) {
        const float* zrow = z + (tok0 + t) * E_DIM;
        for (int d = lane; d < E_DIM; d += 32)
            R[t * LDS_STRIDE + d] = zrow[d];
    }
    __syncthreads();

    float loss_acc = 0.f;

    for (int q = 0; q < N_Q; ++q) {
        const float* Wq = W + (size_t)q * N_E * E_DIM;
        const float* Nq = norms + q * N_E;

        float bv0[8], bv1[8];
        int   bi0[8], bi1[8];
        #pragma unroll
        for (int r = 0; r < 8; ++r) {
            bv0[r] = 3.0e38f; bv1[r] = 3.0e38f;
            bi0[r] = 0;       bi1[r] = 0;
        }

        // ---- distance GEMM over all 2048 codes, 16 codes per chunk ----
        for (int cb = 0; cb < N_E; cb += 16) {
            const int   code = cb + l16;
            const float nrm  = Nq[code];
            // B frag (= Wq^T): lane holds W[code][kk], W[code][kk+1], kk = k+2*half
            const float* wrow = Wq + (size_t)code * E_DIM + 2 * half;
            // A frags: lane holds R[m][kk], R[m][kk+1], m = l16 (+16 for tile 1)
            const float* a0p = R + l16 * LDS_STRIDE + 2 * half;
            const float* a1p = R + (16 + l16) * LDS_STRIDE + 2 * half;

            v8f d0 = {0.f, 0.f, 0.f, 0.f, 0.f, 0.f, 0.f, 0.f};
            v8f d1 = d0;
            #pragma unroll 8
            for (int kk = 0; kk < E_DIM; kk += 4) {
                v2f b  = *(const v2f*)(wrow + kk);
                v2f a0 = *(const v2f*)(a0p + kk);
                v2f a1 = *(const v2f*)(a1p + kk);
                d0 = __builtin_amdgcn_wmma_f32_16x16x4_f32(
                        false, a0, false, b, (short)0, d0, false, false);
                d1 = __builtin_amdgcn_wmma_f32_16x16x4_f32(
                        false, a1, false, b, (short)0, d1, false, false);
            }
            // score = ||code||^2 - 2*dot ; running per-lane argmin
            #pragma unroll
            for (int r = 0; r < 8; ++r) {
                float s0 = fmaf(-2.f, d0[r], nrm);
                if (s0 < bv0[r]) { bv0[r] = s0; bi0[r] = code; }
                float s1 = fmaf(-2.f, d1[r], nrm);
                if (s1 < bv1[r]) { bv1[r] = s1; bi1[r] = code; }
            }
        }

        // ---- cross-lane argmin within each 16-lane half (butterfly) ----
        #pragma unroll
        for (int r = 0; r < 8; ++r) {
            #pragma unroll
            for (int m = 1; m < 16; m <<= 1) {
                float ov = __shfl_xor(bv0[r], m, 16);
                int   oi = __shfl_xor(bi0[r], m, 16);
                if (ov < bv0[r] || (ov == bv0[r] && oi < bi0[r])) { bv0[r] = ov; bi0[r] = oi; }
                ov = __shfl_xor(bv1[r], m, 16);
                oi = __shfl_xor(bi1[r], m, 16);
                if (ov < bv1[r] || (ov == bv1[r] && oi < bi1[r])) { bv1[r] = ov; bi1[r] = oi; }
            }
            if (l16 == 0) {                       // token = r + 8*half (+16 tile 1)
                widx[r + 8 * half]      = bi0[r];
                widx[16 + r + 8 * half] = bi1[r];
            }
        }
        __syncthreads();

        // ---- gather winning code, update residual, accumulate loss ----
        for (int t = 0; t < TPW; ++t) {
            const int ci = widx[t];
            const float* crow = Wq + (size_t)ci * E_DIM;
            float* rrow = R + t * LDS_STRIDE;
            for (int d = lane; d < E_DIM; d += 32) {
                float nr = rrow[d] - crow[d];     // new residual == diff
                rrow[d] = nr;
                loss_acc = fmaf(nr, nr, loss_acc);
            }
            if (lane == 0)
                idx_out[(tok0 + t) * N_Q + q] = (float)ci;
        }
        __syncthreads();
    }

    // z_q = cumulative_q = z - final residual
    for (int t = 0; t < TPW; ++t) {
        const size_t off = (tok0 + t) * E_DIM;
        for (int d = lane; d < E_DIM; d += 32)
            zq[off + d] = z[off + d] - R[t * LDS_STRIDE + d];
    }

    // deterministic per-wave loss partial
    #pragma unroll
    for (int m = 16; m >= 1; m >>= 1) loss_acc += __shfl_xor(loss_acc, m, 32);
    if (lane == 0) loss_partial[tile] = loss_acc;
}

// ---------------------------------------------------------------------------
// Kernel 3: deterministic fixed-order reduction of loss partials.
// loss = (1+beta)/(E_DIM * N_Q * N_TOK) * sum(diff^2)
// ---------------------------------------------------------------------------
__global__ void rvq_loss_reduce_kernel(const float* __restrict__ partials,
                                       float* __restrict__ loss_out) {
    __shared__ float sm[256];
    float s = 0.f;
    for (int i = threadIdx.x; i < NUM_TILES; i += 256) s += partials[i];
    sm[threadIdx.x] = s;
    __syncthreads();
    for (int w = 128; w >= 1; w >>= 1) {
        if ((int)threadIdx.x < w) sm[threadIdx.x] += sm[threadIdx.x + w];
        __syncthreads();
    }
    if (threadIdx.x == 0)
        *loss_out = sm[0] * (1.25f / ((float)E_DIM * (float)N_Q * (float)N_TOK));
}

// ---------------------------------------------------------------------------
extern "C" void kernel_launch(void* const* d_in, const int* in_sizes, int n_in,
                              void* d_out, int out_size, void* d_ws, size_t ws_size,
                              hipStream_t stream) {
    (void)in_sizes; (void)n_in; (void)out_size; (void)ws_size;
    const float* z = (const float*)d_in[0];   // [16,2048,256]
    const float* W = (const float*)d_in[1];   // [4,2048,256]

    float* out      = (float*)d_out;
    float* zq       = out;                    // ZQ_SIZE floats
    float* loss     = out + ZQ_SIZE;          // 1 float
    float* idx_out  = out + ZQ_SIZE + 1;      // N_TOK*N_Q "indices" as float

    float* ws           = (float*)d_ws;
    float* loss_partial = ws;                 // NUM_TILES floats
    float* norms        = ws + NUM_TILES;     // N_Q*N_E floats

    // 1) code norms
    rvq_norms_kernel<<<(N_Q * N_E) / 8, 256, 0, stream>>>(W, norms);

    // 2) main RVQ (dynamic LDS: residual tiles + winner indices)
    const size_t shmem = (size_t)WPW * TPW * LDS_STRIDE * sizeof(float)
                       + (size_t)WPW * TPW * sizeof(int);
    rvq_main_kernel<<<NUM_TILES / WPW, WPW * 32, shmem, stream>>>(
        z, W, norms, zq, idx_out, loss_partial);

    // 3) deterministic loss reduction
    rvq_loss_reduce_kernel<<<1, 256, 0, stream>>>(loss_partial, loss);
}